// Normalized_Graph_expand_37709812859473
// MI455X (gfx1250) — compile-verified
//
#include <hip/hip_runtime.h>
#include <hip/hip_bf16.h>

// Problem shape (fixed by the reference): B=1, N=25000 nodes, F=128 features,
// CUT=32 neighbors per node.  out[i][k][:] = x[g[i][k]][:] - x[i][:]
#define NNODES 25000
#define NCUT   32
#define NF     128

typedef float        v4f __attribute__((ext_vector_type(4)));
typedef unsigned int v4u __attribute__((ext_vector_type(4)));
typedef int          v4i __attribute__((ext_vector_type(4)));
typedef int          v8i __attribute__((ext_vector_type(8)));

typedef __attribute__((address_space(3))) float lds_float;

__global__ __launch_bounds__(256)
void Normalized_Graph_expand_37709812859473_kernel(
    const float* __restrict__ xf,   // [N, F] features (B=1 squeezed)
    const int*   __restrict__ gg,   // [N, CUT] neighbor indices
    float*       __restrict__ out)  // [N, CUT, F]
{
    // One node per block. 16 KB staging tile: 32 neighbor rows x 512 B.
    __shared__ float tile[NCUT * NF];

    const int i    = blockIdx.x;
    const int tid  = threadIdx.x;
    const int lane = tid & 31;
    const int wid  = tid >> 5;

    // ---- TDM gather: wave 0 issues two gather-mode tensor_load_to_lds ----
    // (TENSOR ops ignore EXEC, but this wave-uniform branch lowers to
    //  s_cbranch_execz that jumps over the op for waves 1..7.)
    if (wid == 0) {
        // LDS byte offset of `tile` via addrspace(3) ptrtoint.
        unsigned lds_base =
            (unsigned)(unsigned long long)(lds_float*)&tile[0];

        // Pack 32 neighbor indices into 16 x 16-bit pairs (N < 2^16).
        const int* gi = gg + (long long)i * NCUT;
        unsigned idxw[16];
#pragma unroll
        for (int j = 0; j < 16; ++j) {
            unsigned a = (unsigned)gi[2 * j]     & 0xFFFFu;
            unsigned b = (unsigned)gi[2 * j + 1] & 0xFFFFu;
            idxw[j] = a | (b << 16);
        }

        unsigned long long gaddr = (unsigned long long)(const void*)xf;

        // D# group 0: count=1, gather_mode=1 (16-bit indices),
        // lds_addr, global_addr[56:0], type=2 ("image") in bits [127:126].
        v4u g0a = { 0x80000001u,
                    lds_base,
                    (unsigned)(gaddr & 0xFFFFFFFFu),
                    ((unsigned)((gaddr >> 32) & 0x01FFFFFFu)) | (2u << 30) };
        v4u g0b = g0a;
        g0b.y   = lds_base + 16u * NF * 4u;   // second 8 KB half of the tile

        // D# group 1:
        //  w0: workgroup_mask=0, data_size=2 (4B)         -> 0x00020000
        //  w1: atomic_barrier_addr=0, tensor_dim0=128     -> 128<<16
        //  w2: tensor_dim1=25000 (row-index OOB bound)    -> 25000<<16
        //  w3: tile_dim0=128 (row width)                  -> 128<<16
        //  w4: tile_dim1=16 (#valid gather indices)       -> 16
        //  w5: tensor_dim0_stride=128                     -> 128
        //  w6,w7: stride hi / dim1 stride (ignored in gather mode) = 0
        v8i g1 = { (int)0x00020000u,
                   (int)(128u   << 16),
                   (int)(25000u << 16),
                   (int)(128u   << 16),
                   16, 128, 0, 0 };

        // D# groups 2/3: packed 16-bit row indices (8 words per op).
        v4i g2a = { (int)idxw[0],  (int)idxw[1],  (int)idxw[2],  (int)idxw[3]  };
        v4i g3a = { (int)idxw[4],  (int)idxw[5],  (int)idxw[6],  (int)idxw[7]  };
        v4i g2b = { (int)idxw[8],  (int)idxw[9],  (int)idxw[10], (int)idxw[11] };
        v4i g3b = { (int)idxw[12], (int)idxw[13], (int)idxw[14], (int)idxw[15] };

        // Inline asm bypasses the (toolchain-divergent) clang builtin and maps
        // directly to the VIMAGE encoding: VADDR0=group0 (4 SGPRs),
        // VADDR1=group1 (8 SGPRs), VADDR2/3=groups 2/3 (4 SGPRs each).
        asm volatile("tensor_load_to_lds %0, %1, %2, %3"
                     :
                     : "s"(g0a), "s"(g1), "s"(g2a), "s"(g3a)
                     : "memory");
        asm volatile("tensor_load_to_lds %0, %1, %2, %3"
                     :
                     : "s"(g0b), "s"(g1), "s"(g2b), "s"(g3b)
                     : "memory");
        __builtin_amdgcn_s_wait_tensorcnt(0);
    }
    __syncthreads();

    // ---- center row: one float4 per lane (hits L2; 12.8 MB table is hot) ----
    const v4f* xf4 = (const v4f*)xf;
    v4f c = xf4[(long long)i * (NF / 4) + lane];

    // ---- subtract + stream out with non-temporal b128 stores ----
    v4f*       out4 = (v4f*)out;
    const v4f* t4   = (const v4f*)tile;   // ds_load_b128
#pragma unroll
    for (int r = 0; r < 4; ++r) {
        const int k = wid + r * 8;        // neighbor slot 0..31, one per wave
        v4f nb = t4[k * (NF / 4) + lane];
        v4f d  = nb - c;
        __builtin_nontemporal_store(
            d, &out4[((long long)i * NCUT + k) * (NF / 4) + lane]);
    }
}

extern "C" void kernel_launch(void* const* d_in, const int* in_sizes, int n_in,
                              void* d_out, int out_size, void* d_ws, size_t ws_size,
                              hipStream_t stream) {
    (void)in_sizes; (void)n_in; (void)d_ws; (void)ws_size; (void)out_size;
    const float* xf = (const float*)d_in[0];   // [1, 25000, 128] f32
    const int*   gg = (const int*)d_in[1];     // [1, 25000, 32] i32
    float*       out = (float*)d_out;          // [1, 25000, 32, 128] f32

    dim3 grid(NNODES);
    dim3 block(256);
    Normalized_Graph_expand_37709812859473_kernel<<<grid, block, 0, stream>>>(xf, gg, out);
}